// MaskedUncertaintyChamferLoss_57638461112904
// MI455X (gfx1250) — compile-verified
//
#include <hip/hip_runtime.h>
#include <math.h>

// CDNA5 / gfx1250: wave32, WMMA f32 16x16x4 for the K=3 cross-term GEMM.

typedef __attribute__((ext_vector_type(2))) float v2f;
typedef __attribute__((ext_vector_type(8))) float v8f;

#define ALPHA 1.0f
#define FLT_BIG 3.0e38f

// min across each 16-lane half (lanes 0-15 and 16-31 reduced independently)
__device__ __forceinline__ float half_min16(float v) {
  v = fminf(v, __shfl_xor(v, 1, 32));
  v = fminf(v, __shfl_xor(v, 2, 32));
  v = fminf(v, __shfl_xor(v, 4, 32));
  v = fminf(v, __shfl_xor(v, 8, 32));
  return v;
}

__device__ __forceinline__ float wave_sum32(float v) {
  v += __shfl_xor(v, 16, 32);
  v += __shfl_xor(v, 8, 32);
  v += __shfl_xor(v, 4, 32);
  v += __shfl_xor(v, 2, 32);
  v += __shfl_xor(v, 1, 32);
  return v;
}

// select a[lane&7] without dynamic indexing (keeps everything in VGPRs)
__device__ __forceinline__ float select_r(const float* a, int r) {
  float v = a[0];
  #pragma unroll
  for (int j = 1; j < 8; ++j) v = (r == j) ? a[j] : v;
  return v;
}

// --------------------------------------------------------------------------
// pred -> gt direction: one wave owns a 16-pred tile, streams all gt tiles.
// Cross tile D[m=pred][n=gt] via v_wmma_f32_16x16x4_f32 (K padded 3->4).
// A layout (16x4 f32): lanes 0-15 hold {K0,K1}, lanes 16-31 hold {K2,K3},
// M = lane&15. B layout mirrored: N = lane&15, lane half selects K pair.
// Hoisted out of the hot loop (both commute with min over n):
//   * the row norm pn[m] (constant per output row) -> added after the loop
//   * the clamp max(0,.) (monotone)                -> applied after the loop
// Inner loop per tile: b96 load, col norm, 1 WMMA, 4 pk_fma, merged min3.
// --------------------------------------------------------------------------
__global__ __launch_bounds__(256) void p2g_kernel(
    const float* __restrict__ x_gt, const float* __restrict__ x_pred,
    const int* __restrict__ mask, const float* __restrict__ conf,
    float* __restrict__ sum_out, int B, int V1, int V2) {
  const int lane = threadIdx.x & 31;
  const int wave = (int)((blockIdx.x * blockDim.x + threadIdx.x) >> 5);
  const int tilesPerB = V2 >> 4;
  const int b    = wave / tilesPerB;
  const int tile = wave % tilesPerB;
  const int nl   = lane & 15;
  const bool hi  = lane >= 16;

  // Stationary A operand: this wave's 16 pred points.
  const float* p = x_pred + ((size_t)b * V2 + (size_t)tile * 16 + nl) * 3;
  const float px = p[0], py = p[1], pz = p[2];
  const float pnl = px * px + py * py + pz * pz;   // |pred(nl)|^2
  v2f Av;
  Av.x = hi ? pz : px;   // K2 : K0
  Av.y = hi ? 0.f : py;  // K3 : K1   (zero-pad K=3)
  float pn_r[8];
  #pragma unroll
  for (int r = 0; r < 8; ++r) pn_r[r] = __shfl(pnl, r + (hi ? 8 : 0), 32);

  float dmin[8];          // running min of (gn - 2*cross); pn added at end
  #pragma unroll
  for (int r = 0; r < 8; ++r) dmin[r] = FLT_BIG;

  const float* gbase = x_gt + (size_t)b * V1 * 3;
  const int ntiles = V1 >> 4;
  #pragma unroll 4
  for (int t = 0; t < ntiles; ++t) {
    const float* g = gbase + ((size_t)t * 16 + nl) * 3;
    __builtin_prefetch(g + 192, 0, 3);             // ~4 tiles ahead
    const float gx = g[0], gy = g[1], gz = g[2];
    const float gn = gx * gx + gy * gy + gz * gz;  // |gt(nl)|^2
    v2f Bv;
    Bv.x = hi ? gz : gx;
    Bv.y = hi ? 0.f : gy;
    v8f D = {};
    D = __builtin_amdgcn_wmma_f32_16x16x4_f32(false, Av, false, Bv,
                                              (short)0, D, false, false);
    #pragma unroll
    for (int r = 0; r < 8; ++r)
      dmin[r] = fminf(dmin[r], fmaf(-2.f, D[r], gn));
  }

  #pragma unroll
  for (int r = 0; r < 8; ++r) dmin[r] = half_min16(dmin[r]) + pn_r[r];

  // lanes 0-7 carry m=0..7, lanes 16-23 carry m=8..15
  float contrib = 0.f;
  if (nl < 8) {
    const int r = lane & 7;
    const float dm = fmaxf(select_r(dmin, r), 0.f);   // hoisted clamp
    const int m = r + (hi ? 8 : 0);
    const size_t pidx = (size_t)b * V2 + (size_t)tile * 16 + m;
    const float c = conf[pidx];
    if (mask[pidx] > 0) contrib = dm * c - ALPHA * logf(c);
  }
  contrib = wave_sum32(contrib);
  if (lane == 0) atomicAdd(sum_out, contrib);
}

// --------------------------------------------------------------------------
// gt -> pred direction: one wave owns a 16-gt tile, streams all pred tiles.
// Invalid preds are masked by poisoning the pred norm with FLT_BIG: the
// column value (pn' - 2*cross) stays ~FLT_BIG (float granularity at 3e38
// swallows the bounded cross term), which acts as +inf for the min. The gt
// row norm gn[m] is hoisted out of the loop like pn in p2g.
// --------------------------------------------------------------------------
__global__ __launch_bounds__(256) void g2p_kernel(
    const float* __restrict__ x_gt, const float* __restrict__ x_pred,
    const int* __restrict__ mask,
    float* __restrict__ sum_out, int B, int V1, int V2) {
  const int lane = threadIdx.x & 31;
  const int wave = (int)((blockIdx.x * blockDim.x + threadIdx.x) >> 5);
  const int tilesPerB = V1 >> 4;
  const int b    = wave / tilesPerB;
  const int tile = wave % tilesPerB;
  const int nl   = lane & 15;
  const bool hi  = lane >= 16;

  const float* g = x_gt + ((size_t)b * V1 + (size_t)tile * 16 + nl) * 3;
  const float gx = g[0], gy = g[1], gz = g[2];
  const float gnl = gx * gx + gy * gy + gz * gz;
  v2f Av;
  Av.x = hi ? gz : gx;
  Av.y = hi ? 0.f : gy;
  float gn_r[8];
  #pragma unroll
  for (int r = 0; r < 8; ++r) gn_r[r] = __shfl(gnl, r + (hi ? 8 : 0), 32);

  float dmin[8];          // running min of (pn' - 2*cross); gn added at end
  #pragma unroll
  for (int r = 0; r < 8; ++r) dmin[r] = FLT_BIG;

  const float* pbase = x_pred + (size_t)b * V2 * 3;
  const int* mbase = mask + (size_t)b * V2;
  const int ntiles = V2 >> 4;
  #pragma unroll 4
  for (int t = 0; t < ntiles; ++t) {
    const int pidx = t * 16 + nl;
    const float* p = pbase + (size_t)pidx * 3;
    __builtin_prefetch(p + 192, 0, 3);
    const float px = p[0], py = p[1], pz = p[2];
    float pn = px * px + py * py + pz * pz;
    pn = (mbase[pidx] > 0) ? pn : FLT_BIG;         // mask folded into norm
    v2f Bv;
    Bv.x = hi ? pz : px;
    Bv.y = hi ? 0.f : py;
    v8f D = {};
    D = __builtin_amdgcn_wmma_f32_16x16x4_f32(false, Av, false, Bv,
                                              (short)0, D, false, false);
    #pragma unroll
    for (int r = 0; r < 8; ++r)
      dmin[r] = fminf(dmin[r], fmaf(-2.f, D[r], pn));
  }

  #pragma unroll
  for (int r = 0; r < 8; ++r) dmin[r] = half_min16(dmin[r]) + gn_r[r];

  float contrib = 0.f;
  if (nl < 8) {
    const int r = lane & 7;
    contrib = fmaxf(select_r(dmin, r), 0.f);  // d_g2p for gt tile*16+r+8*hi
  }
  contrib = wave_sum32(contrib);
  if (lane == 0) atomicAdd(sum_out, contrib);
}

// --------------------------------------------------------------------------
__global__ void init_kernel(float* __restrict__ sums, int* __restrict__ cnts,
                            int B) {
  const int i = threadIdx.x;
  if (i < 2) sums[i] = 0.f;
  if (i < B) cnts[i] = 0;
}

__global__ void count_kernel(const int* __restrict__ mask,
                             int* __restrict__ cnts, int V2) {
  const int i = (int)(blockIdx.x * blockDim.x + threadIdx.x);
  const int b = i / V2;                       // V2 % 32 == 0 -> wave-uniform
  const int valid = (mask[i] > 0) ? 1 : 0;
  const unsigned long long bal = __ballot(valid);
  if ((threadIdx.x & 31) == 0) atomicAdd(&cnts[b], (int)__popcll(bal));
}

__global__ void finalize_kernel(const float* __restrict__ sums,
                                const int* __restrict__ cnts,
                                float* __restrict__ out, int B, int V1) {
  if (threadIdx.x == 0 && blockIdx.x == 0) {
    int mc = 0;
    for (int b = 0; b < B; ++b) mc = mc > cnts[b] ? mc : cnts[b];
    const float loss_p2g = sums[0] / ((float)B * (float)mc);
    const float loss_g2p = sums[1] / ((float)B * (float)V1);
    out[0] = loss_p2g + loss_g2p;
  }
}

// --------------------------------------------------------------------------
extern "C" void kernel_launch(void* const* d_in, const int* in_sizes, int n_in,
                              void* d_out, int out_size, void* d_ws,
                              size_t ws_size, hipStream_t stream) {
  const float* x_gt   = (const float*)d_in[0];  // (B,V1,3) f32
  const float* x_pred = (const float*)d_in[1];  // (B,V2,3) f32
  const int*   mask   = (const int*)d_in[2];    // (B,V2,1) i32
  const float* conf   = (const float*)d_in[3];  // (B,V2)   f32
  float* out = (float*)d_out;

  const int B  = 4;
  const int V1 = in_sizes[0] / (B * 3);
  const int V2 = in_sizes[1] / (B * 3);

  float* sums = (float*)d_ws;              // [0]=p2g sum, [1]=g2p sum
  int*   cnts = (int*)(sums + 2);          // per-batch valid counts

  init_kernel<<<1, 64, 0, stream>>>(sums, cnts, B);
  count_kernel<<<(B * V2) / 256, 256, 0, stream>>>(mask, cnts, V2);

  const int wavesP = (B * V2) / 16;        // one wave per 16-pred tile
  p2g_kernel<<<wavesP / 8, 256, 0, stream>>>(x_gt, x_pred, mask, conf,
                                             &sums[0], B, V1, V2);
  const int wavesG = (B * V1) / 16;        // one wave per 16-gt tile
  g2p_kernel<<<wavesG / 8, 256, 0, stream>>>(x_gt, x_pred, mask,
                                             &sums[1], B, V1, V2);
  finalize_kernel<<<1, 32, 0, stream>>>(sums, cnts, out, B, V1);
}